// CGSLayer_36361193127946
// MI455X (gfx1250) — compile-verified
//
#include <hip/hip_runtime.h>
#include <math.h>

#define N_NODES 100000
#define IN_W    64
#define OUT_W   64
#define N_EDGES 1600000

typedef __attribute__((ext_vector_type(2))) float v2f;
typedef __attribute__((ext_vector_type(8))) float v8f;

// ---------------------------------------------------------------------------
// Phase 1: degree / normalization  dinv[i] = rsqrt(2.0 + sum_{e: dst==i} w[e])
// ---------------------------------------------------------------------------
__global__ void deg_init_kernel(float* __restrict__ deg) {
    int i = blockIdx.x * blockDim.x + threadIdx.x;
    if (i < N_NODES) deg[i] = 2.0f;             // self-loop weight (improved=True)
}

__global__ void deg_accum_kernel(const int* __restrict__ dst,
                                 const float* __restrict__ w,
                                 float* __restrict__ deg) {
    int e = blockIdx.x * blockDim.x + threadIdx.x;
    if (e < N_EDGES) atomicAdd(&deg[dst[e]], w[e]);
}

__global__ void deg_rsqrt_kernel(float* __restrict__ deg) {
    int i = blockIdx.x * blockDim.x + threadIdx.x;
    if (i < N_NODES) {
        float d = deg[i];
        deg[i] = (d > 0.0f) ? rsqrtf(d) : 0.0f; // becomes dinv[] in place
    }
}

// ---------------------------------------------------------------------------
// Phase 2: h = x @ W  via V_WMMA_F32_16X16X4_F32 (f32 matrix core path)
// One wave32 per 16x16 output tile; K=64 in 16 steps of 4.
// A 16x4 f32 layout: lanes 0-15 -> K={0,1}, lanes 16-31 -> K={2,3}, M=lane%16.
// B 4x16 f32 layout mirrors A with N=lane%16.
// C/D: VGPR v holds row (v + 8*(lane>=16)), col lane%16.
// ---------------------------------------------------------------------------
__global__ void gemm_wmma_kernel(const float* __restrict__ x,
                                 const float* __restrict__ W,
                                 float* __restrict__ h) {
    const unsigned lane  = threadIdx.x & 31u;
    const unsigned gwave = blockIdx.x * 8u + (threadIdx.x >> 5);
    const unsigned tile_n = gwave & 3u;          // 64/16 = 4 N-tiles
    const unsigned tile_m = gwave >> 2;          // 100000/16 = 6250 M-tiles
    const unsigned m0 = tile_m * 16u;
    const unsigned n0 = tile_n * 16u;
    const unsigned half = lane >> 4;             // 0: K lo pair, 1: K hi pair
    const unsigned lrow = lane & 15u;

    const float* xrow = x + (size_t)(m0 + lrow) * IN_W;   // A: row of x
    const float* wcol = W + (n0 + lrow);                  // B: column of W

    v8f c = {};
#pragma unroll
    for (int kk = 0; kk < 16; ++kk) {
        const int kb = kk * 4 + (int)half * 2;            // even -> b64 aligned
        v2f a = *(const v2f*)(xrow + kb);                  // x[m][kb], x[m][kb+1]
        v2f b;
        b.x = wcol[(size_t)kb * OUT_W];                    // W[kb  ][n]
        b.y = wcol[(size_t)(kb + 1) * OUT_W];              // W[kb+1][n]
        c = __builtin_amdgcn_wmma_f32_16x16x4_f32(
                /*neg_a=*/false, a, /*neg_b=*/false, b,
                /*c_mod=*/(short)0, c, /*reuse_a=*/false, /*reuse_b=*/false);
    }

    float* hcol = h + n0 + lrow;
#pragma unroll
    for (int v = 0; v < 8; ++v) {
        hcol[(size_t)(m0 + v + half * 8u) * OUT_W] = c[v];
    }
}

// ---------------------------------------------------------------------------
// Phase 3: out[i] = b + h[i] * (dinv[i] * 2.0 * dinv[i])   (self-loop term)
// One thread per 4 features (float4).
// ---------------------------------------------------------------------------
__global__ void out_init_kernel(const float* __restrict__ h,
                                const float* __restrict__ dinv,
                                const float* __restrict__ bias,
                                float* __restrict__ out) {
    int t = blockIdx.x * blockDim.x + threadIdx.x;       // N_NODES*16 threads
    if (t >= N_NODES * (OUT_W / 4)) return;
    int node = t >> 4;
    int f    = (t & 15) * 4;
    float di = dinv[node];
    float s  = 2.0f * di * di;
    float4 hv = *(const float4*)(h + (size_t)node * OUT_W + f);
    float4 bv = *(const float4*)(bias + f);
    float4 o;
    o.x = bv.x + hv.x * s;
    o.y = bv.y + hv.y * s;
    o.z = bv.z + hv.z * s;
    o.w = bv.w + hv.w * s;
    *(float4*)(out + (size_t)node * OUT_W + f) = o;
}

// ---------------------------------------------------------------------------
// Phase 4: edge scatter  out[dst] += h[src] * (dinv[src]*w*dinv[dst])
// 16 lanes per edge, float4 gather + 4x global_atomic_add_f32 per lane.
// This phase is the bandwidth roofline (~1 GB traffic -> ~45us @ 23.3 TB/s).
// ---------------------------------------------------------------------------
__global__ void edge_scatter_kernel(const int* __restrict__ src,
                                    const int* __restrict__ dst,
                                    const float* __restrict__ w,
                                    const float* __restrict__ dinv,
                                    const float* __restrict__ h,
                                    float* __restrict__ out) {
    int t = blockIdx.x * blockDim.x + threadIdx.x;       // N_EDGES*16 threads
    int e = t >> 4;
    if (e >= N_EDGES) return;
    int f = (t & 15) * 4;
    int s = src[e];
    int d = dst[e];
    float coef = dinv[s] * w[e] * dinv[d];
    float4 hv = *(const float4*)(h + (size_t)s * OUT_W + f);
    float* o = out + (size_t)d * OUT_W + f;
    atomicAdd(o + 0, hv.x * coef);
    atomicAdd(o + 1, hv.y * coef);
    atomicAdd(o + 2, hv.z * coef);
    atomicAdd(o + 3, hv.w * coef);
}

// ---------------------------------------------------------------------------
extern "C" void kernel_launch(void* const* d_in, const int* in_sizes, int n_in,
                              void* d_out, int out_size, void* d_ws, size_t ws_size,
                              hipStream_t stream) {
    const float* x         = (const float*)d_in[0];   // [N, 64]
    const float* edge_attr = (const float*)d_in[1];   // [E]
    const float* W         = (const float*)d_in[2];   // [64, 64]
    const float* bias      = (const float*)d_in[3];   // [64]
    const int*   edge_idx  = (const int*)d_in[4];     // [2, E]
    // d_in[5] = batch (unused by the reference math)
    float* out = (float*)d_out;                       // [N, 64]

    // Workspace layout: dinv (N f32, padded) | h (N*64 f32)
    float* dinv = (float*)d_ws;
    float* h    = dinv + 100096;                      // 128-aligned past N_NODES

    const int* src = edge_idx;                        // edge_index[0]
    const int* dst = edge_idx + N_EDGES;              // edge_index[1]

    // Phase 1: degrees -> dinv
    deg_init_kernel<<<(N_NODES + 255) / 256, 256, 0, stream>>>(dinv);
    deg_accum_kernel<<<(N_EDGES + 255) / 256, 256, 0, stream>>>(dst, edge_attr, dinv);
    deg_rsqrt_kernel<<<(N_NODES + 255) / 256, 256, 0, stream>>>(dinv);

    // Phase 2: h = x @ W  (6250 M-tiles * 4 N-tiles = 25000 waves = 3125 blocks)
    gemm_wmma_kernel<<<3125, 256, 0, stream>>>(x, W, h);

    // Phase 3: init out with bias + self-loop contribution
    out_init_kernel<<<(N_NODES * 16 + 255) / 256, 256, 0, stream>>>(h, dinv, bias, out);

    // Phase 4: edge scatter-add
    edge_scatter_kernel<<<(N_EDGES * 16 + 255) / 256, 256, 0, stream>>>(
        src, dst, edge_attr, dinv, h, out);
}